// CHIVE_25426206392931
// MI455X (gfx1250) — compile-verified
//
#include <hip/hip_runtime.h>
#include <math.h>

// Clockwork RNN (CHIVE) for MI455X / gfx1250.
// T=2048 steps sequential, B=2048 rows parallel, H=32.
// One wave32 owns 16 batch rows for the entire sequence; weights live in
// per-lane WMMA B-fragments (registers), h-states live in wave-private LDS
// (row-major) so WMMA C/D output can be re-read in A layout next step.
// Matrix op: V_WMMA_F32_16X16X4_F32 (native fp32, matches reference numerics).
// tanh: native v_tanh_f32. Bias folded into accumulator init (bias is
// constant per lane in C/D layout since N = lane&15 across all 8 VGPRs).

#define HID 32
#define TT  2048
#define BB  2048
#define DF  8
#define DP  8
#define DSY 24
#define WAVES_PER_BLOCK 4
#define ROWS_PER_WAVE   16
#define STATE_PER_WAVE  (5 * ROWS_PER_WAVE * HID)   // hf, hp, hs[3]

typedef __attribute__((ext_vector_type(2))) float v2f;
typedef __attribute__((ext_vector_type(8))) float v8f;

__device__ __forceinline__ v8f wmma_f32(v2f a, v2f b, v8f c) {
    // D = A(16x4,f32) * B(4x16,f32) + C(16x16,f32)
    return __builtin_amdgcn_wmma_f32_16x16x4_f32(
        /*neg_a=*/false, a, /*neg_b=*/false, b,
        /*c_mod=*/(short)0, c, /*reuse_a=*/false, /*reuse_b=*/false);
}

__device__ __forceinline__ float fast_tanh(float x) {
#if __has_builtin(__builtin_amdgcn_tanhf)
    // gfx1250 native V_TANH_F32 (single TRANS32 op, co-executes with matrix pipe)
    return __builtin_amdgcn_tanhf(x);
#else
    float z = x * 2.885390081777927f;            // 2*log2(e)
    z = fminf(fmaxf(z, -64.0f), 64.0f);
    float e = __builtin_amdgcn_exp2f(z);
    return 1.0f - 2.0f * __builtin_amdgcn_rcpf(e + 1.0f);
#endif
}

__device__ __forceinline__ v8f splat8(float x) {
    return v8f{x, x, x, x, x, x, x, x};
}

__launch_bounds__(WAVES_PER_BLOCK * 32, 1)
__global__ void chive_clockwork_kernel(
    const float* __restrict__ xf,   // [T,B,8]
    const float* __restrict__ xp,   // [T,B,8]
    const float* __restrict__ xs,   // [T,B,24]
    const float* __restrict__ Wxf, const float* __restrict__ Whf, const float* __restrict__ bf,
    const float* __restrict__ Wxp, const float* __restrict__ Whp, const float* __restrict__ bp,
    const float* __restrict__ Wxs, const float* __restrict__ Whs, const float* __restrict__ bs,
    const int* __restrict__ fclk, const int* __restrict__ pclk, const int* __restrict__ sfreq,
    float* __restrict__ out)        // [3,B,32]
{
    __shared__ float lds[WAVES_PER_BLOCK * STATE_PER_WAVE];

    const int lane = threadIdx.x & 31;
    const int wave = threadIdx.x >> 5;
    const int c    = lane & 15;   // N (col) within 16-tile; also D col
    const int half = lane >> 4;   // lane half selects K pair / M+8
    const int r    = lane & 15;   // A row within 16
    const int wrow = (blockIdx.x * WAVES_PER_BLOCK + wave) * ROWS_PER_WAVE;
    const int grow = wrow + r;    // this lane's global batch row for A loads

    float* hf = &lds[wave * STATE_PER_WAVE];
    float* hp = hf + ROWS_PER_WAVE * HID;
    float* hs = hp + ROWS_PER_WAVE * HID;   // 3 stacks, consecutive

    // Zero wave-private state (graph replays: no reliance on prior contents).
    for (int i = lane; i < STATE_PER_WAVE; i += 32) hf[i] = 0.0f;

    // ---- Preload weight B-fragments: b[v] = W[kb*4 + half*2 + v][nt*16 + c]
    v2f fWxf[2][2], fWhf[8][2], fWxp[2][2], fWhp[8][2], fWxs[8][2], fWhs[8][2];
#pragma unroll
    for (int kb = 0; kb < 8; ++kb) {
        const int k0 = kb * 4 + half * 2;
#pragma unroll
        for (int nt = 0; nt < 2; ++nt) {
            const int col = nt * 16 + c;
            fWhf[kb][nt] = v2f{Whf[(k0 + 0) * HID + col], Whf[(k0 + 1) * HID + col]};
            fWhp[kb][nt] = v2f{Whp[(k0 + 0) * HID + col], Whp[(k0 + 1) * HID + col]};
            fWxs[kb][nt] = v2f{Wxs[(k0 + 0) * HID + col], Wxs[(k0 + 1) * HID + col]};
            fWhs[kb][nt] = v2f{Whs[(k0 + 0) * HID + col], Whs[(k0 + 1) * HID + col]};
            if (kb < 2) {
                fWxf[kb][nt] = v2f{Wxf[(k0 + 0) * HID + col], Wxf[(k0 + 1) * HID + col]};
                fWxp[kb][nt] = v2f{Wxp[(k0 + 0) * HID + col], Wxp[(k0 + 1) * HID + col]};
            }
        }
    }
    // Bias splats: in C/D layout every accumulator VGPR of this lane holds
    // column nt*16+c, so the bias is a per-lane constant -> fold into C init.
    const v8f bf0 = splat8(bf[c]),  bf1 = splat8(bf[16 + c]);
    const v8f bp0 = splat8(bp[c]),  bp1 = splat8(bp[16 + c]);
    const v8f bs0 = splat8(bs[c]),  bs1 = splat8(bs[16 + c]);

    for (int t = 0; t < TT; ++t) {
        const int uf = ((t % (fclk[t] + 1)) == 0);
        const int up = ((t % (pclk[t] + 1)) == 0);
        const int us = (sfreq[t] == 1);

        // ---- frnn cell: h_f = tanh(x_f @ Wx_f + h_f @ Wh_f + b_f)
        if (uf) {
            v8f a0 = bf0, a1 = bf1;
            const float* xrow = xf + ((size_t)t * BB + grow) * DF;
#pragma unroll
            for (int kb = 0; kb < 2; ++kb) {
                v2f a = *(const v2f*)(xrow + kb * 4 + half * 2);
                a0 = wmma_f32(a, fWxf[kb][0], a0);
                a1 = wmma_f32(a, fWxf[kb][1], a1);
            }
#pragma unroll
            for (int kb = 0; kb < 8; ++kb) {
                v2f a = *(const v2f*)(hf + r * HID + kb * 4 + half * 2);
                a0 = wmma_f32(a, fWhf[kb][0], a0);
                a1 = wmma_f32(a, fWhf[kb][1], a1);
            }
#pragma unroll
            for (int v = 0; v < 8; ++v) {
                hf[(half * 8 + v) * HID + c]      = fast_tanh(a0[v]);
                hf[(half * 8 + v) * HID + 16 + c] = fast_tanh(a1[v]);
            }
        }

        // ---- phrnn cell
        if (up) {
            v8f a0 = bp0, a1 = bp1;
            const float* xrow = xp + ((size_t)t * BB + grow) * DP;
#pragma unroll
            for (int kb = 0; kb < 2; ++kb) {
                v2f a = *(const v2f*)(xrow + kb * 4 + half * 2);
                a0 = wmma_f32(a, fWxp[kb][0], a0);
                a1 = wmma_f32(a, fWxp[kb][1], a1);
            }
#pragma unroll
            for (int kb = 0; kb < 8; ++kb) {
                v2f a = *(const v2f*)(hp + r * HID + kb * 4 + half * 2);
                a0 = wmma_f32(a, fWhp[kb][0], a0);
                a1 = wmma_f32(a, fWhp[kb][1], a1);
            }
#pragma unroll
            for (int v = 0; v < 8; ++v) {
                hp[(half * 8 + v) * HID + c]      = fast_tanh(a0[v]);
                hp[(half * 8 + v) * HID + 16 + c] = fast_tanh(a1[v]);
            }
        }

        // ---- syl cell over 3 stacks: inputs = {h_f(new), h_p(new), pad(x_s)}
        if (us) {
#pragma unroll
            for (int j = 0; j < 3; ++j) {
                v8f a0 = bs0, a1 = bs1;
                if (j < 2) {
                    const float* hin = (j == 0) ? hf : hp;
#pragma unroll
                    for (int kb = 0; kb < 8; ++kb) {
                        v2f a = *(const v2f*)(hin + r * HID + kb * 4 + half * 2);
                        a0 = wmma_f32(a, fWxs[kb][0], a0);
                        a1 = wmma_f32(a, fWxs[kb][1], a1);
                    }
                } else {
                    // x_s is zero-padded 24 -> 32: K-blocks 6,7 contribute 0, skip.
                    const float* xrow = xs + ((size_t)t * BB + grow) * DSY;
#pragma unroll
                    for (int kb = 0; kb < 6; ++kb) {
                        v2f a = *(const v2f*)(xrow + kb * 4 + half * 2);
                        a0 = wmma_f32(a, fWxs[kb][0], a0);
                        a1 = wmma_f32(a, fWxs[kb][1], a1);
                    }
                }
                float* hsj = hs + j * ROWS_PER_WAVE * HID;
#pragma unroll
                for (int kb = 0; kb < 8; ++kb) {
                    v2f a = *(const v2f*)(hsj + r * HID + kb * 4 + half * 2);
                    a0 = wmma_f32(a, fWhs[kb][0], a0);
                    a1 = wmma_f32(a, fWhs[kb][1], a1);
                }
#pragma unroll
                for (int v = 0; v < 8; ++v) {
                    hsj[(half * 8 + v) * HID + c]      = fast_tanh(a0[v]);
                    hsj[(half * 8 + v) * HID + 16 + c] = fast_tanh(a1[v]);
                }
            }
        }
    }

    // ---- Emit final h_s : out[j, wrow+rr, cc]
    for (int i = lane; i < 3 * ROWS_PER_WAVE * HID; i += 32) {
        const int j   = i / (ROWS_PER_WAVE * HID);
        const int rem = i % (ROWS_PER_WAVE * HID);
        const int rr  = rem / HID;
        const int cc  = rem % HID;
        out[((size_t)j * BB + (wrow + rr)) * HID + cc] = hs[i];
    }
}

extern "C" void kernel_launch(void* const* d_in, const int* in_sizes, int n_in,
                              void* d_out, int out_size, void* d_ws, size_t ws_size,
                              hipStream_t stream) {
    (void)in_sizes; (void)n_in; (void)out_size; (void)d_ws; (void)ws_size;
    const float* xf  = (const float*)d_in[0];
    const float* xp  = (const float*)d_in[1];
    const float* xs  = (const float*)d_in[2];
    const float* Wxf = (const float*)d_in[3];
    const float* Whf = (const float*)d_in[4];
    const float* bf  = (const float*)d_in[5];
    const float* Wxp = (const float*)d_in[6];
    const float* Whp = (const float*)d_in[7];
    const float* bp  = (const float*)d_in[8];
    const float* Wxs = (const float*)d_in[9];
    const float* Whs = (const float*)d_in[10];
    const float* bs  = (const float*)d_in[11];
    const int* fclk  = (const int*)d_in[12];
    const int* pclk  = (const int*)d_in[13];
    const int* sfreq = (const int*)d_in[14];
    float* out = (float*)d_out;

    const int blocks = BB / (WAVES_PER_BLOCK * ROWS_PER_WAVE);  // 32
    chive_clockwork_kernel<<<blocks, WAVES_PER_BLOCK * 32, 0, stream>>>(
        xf, xp, xs, Wxf, Whf, bf, Wxp, Whp, bp, Wxs, Whs, bs,
        fclk, pclk, sfreq, out);
}